// stream_density_wfa_26731876450882
// MI455X (gfx1250) — compile-verified
//
#include <hip/hip_runtime.h>

#define XDIM 64
#define DDIM 512
#define RDIM 512
#define MIXN 8
#define WINN 4096

typedef __attribute__((ext_vector_type(2))) float v2f;
typedef __attribute__((ext_vector_type(8))) float v8f;

// One full wave computes y[row0..row0+16) = W[row0+m, :] . x  (raw, no bias).
// Uses V_WMMA_F32_16X16X4_F32 with B = x broadcast across all 16 N columns,
// so every column of the 16x16 D tile holds the matvec result.
// A 16x4 layout: lane m (0..15), VGPR v: K = v + 2*(lane>=16)
// B 4x16 layout: VGPR0 = rows K0 (lanes 0-15) / K2 (lanes 16-31); VGPR1 = K1/K3
__device__ __forceinline__ void wave_matvec16(const float* __restrict__ W, int ldw,
                                              int row0, int K,
                                              const float* __restrict__ x,
                                              float* __restrict__ y) {
  const int lane = threadIdx.x & 31;
  const int m = lane & 15;
  const int kh = (lane >> 4) << 1;  // 0 for lanes 0-15, 2 for lanes 16-31
  const float* wrow = W + (size_t)(row0 + m) * (size_t)ldw;
  v8f acc = {0.f, 0.f, 0.f, 0.f, 0.f, 0.f, 0.f, 0.f};
  for (int k0 = 0; k0 < K; k0 += 4) {
    v2f a, b;
    a[0] = wrow[k0 + kh];
    a[1] = wrow[k0 + kh + 1];
    b[0] = x[k0 + kh];          // broadcast across N columns
    b[1] = x[k0 + kh + 1];
    acc = __builtin_amdgcn_wmma_f32_16x16x4_f32(false, a, false, b,
                                                (short)0, acc, false, false);
  }
  // Extract column N=0: lane 0 holds M=0..7 in VGPRs 0..7; lane 16 holds M=8..15.
  if (lane == 0) {
#pragma unroll
    for (int i = 0; i < 8; ++i) y[row0 + i] = acc[i];
  } else if (lane == 16) {
#pragma unroll
    for (int i = 0; i < 8; ++i) y[row0 + 8 + i] = acc[i];
  }
}

// Kernel 1: e = relu(W2 @ relu(W1 @ x + b1) + b2), x = X[WIN-2]
__global__ __launch_bounds__(256) void k_prep(const float* __restrict__ X,
                                              const float* __restrict__ W1,
                                              const float* __restrict__ b1,
                                              const float* __restrict__ W2,
                                              const float* __restrict__ b2,
                                              float* __restrict__ evec) {
  __shared__ float xs[XDIM];
  __shared__ float e1[DDIM];
  __shared__ float e2[DDIM];
  const int t = threadIdx.x;
  if (t < XDIM) xs[t] = X[(size_t)(WINN - 2) * XDIM + t];
  __syncthreads();
  for (int j = t; j < DDIM; j += 256) {
    float s = b1[j];
    const float* w = W1 + (size_t)j * XDIM;
#pragma unroll 8
    for (int k = 0; k < XDIM; ++k) s = fmaf(w[k], xs[k], s);
    e1[j] = s > 0.f ? s : 0.f;
  }
  __syncthreads();
  const int wave = t >> 5;
  for (int tile = wave; tile < DDIM / 16; tile += 8)
    wave_matvec16(W2, DDIM, tile * 16, DDIM, e1, e2);
  __syncthreads();
  for (int j = t; j < DDIM; j += 256) {
    float s = e2[j] + b2[j];
    evec[j] = s > 0.f ? s : 0.f;
  }
}

// Kernel 2: stream all of A (537 MB), partials[blk][j] = sum over block's rows r
// of prev[r>>9]*e[r&511] * A[r][j].  Coalesced b128 loads, 4 cols per thread.
__global__ __launch_bounds__(256) void k_reduce(const float* __restrict__ A,
                                                const float* __restrict__ prev,
                                                const float* __restrict__ evec,
                                                float* __restrict__ partials,
                                                int rowsPerBlock) {
  __shared__ float wl[4096];          // per-row weights (rowsPerBlock <= 4096)
  __shared__ float4 red[128];
  const int t = threadIdx.x;
  const long r0 = (long)blockIdx.x * (long)rowsPerBlock;
  for (int idx = t; idx < rowsPerBlock; idx += 256) {
    long r = r0 + idx;
    wl[idx] = prev[r >> 9] * evec[r & 511];
  }
  __syncthreads();
  const int c = t & 127;              // column group: floats [4c, 4c+4)
  const int p = t >> 7;               // row phase 0/1
  const float4* A4 = reinterpret_cast<const float4*>(A + r0 * 512) + c;
  float4 acc = make_float4(0.f, 0.f, 0.f, 0.f);
#pragma unroll 4
  for (int r = p; r < rowsPerBlock; r += 2) {
    const float w = wl[r];
    const float4 v = A4[(size_t)r * 128];
    acc.x = fmaf(w, v.x, acc.x);
    acc.y = fmaf(w, v.y, acc.y);
    acc.z = fmaf(w, v.z, acc.z);
    acc.w = fmaf(w, v.w, acc.w);
  }
  if (p == 1) red[c] = acc;
  __syncthreads();
  if (p == 0) {
    const float4 o = red[c];
    acc.x += o.x; acc.y += o.y; acc.z += o.z; acc.w += o.w;
    float4* out = reinterpret_cast<float4*>(partials + (size_t)blockIdx.x * DDIM) + c;
    *out = acc;
  }
}

// Kernel 3: h = sigmoid(sum partials); epilogue matvecs (WMMA) + mixture math.
__global__ __launch_bounds__(256) void k_final(const float* __restrict__ X,
                                               const float* __restrict__ partials, int NB,
                                               const float* __restrict__ Wmu,
                                               const float* __restrict__ bmu,
                                               const float* __restrict__ Wmu2,
                                               const float* __restrict__ bmu2,
                                               const float* __restrict__ Wsig,
                                               const float* __restrict__ bsig,
                                               const float* __restrict__ Wal,
                                               const float* __restrict__ bal,
                                               float* __restrict__ out) {
  __shared__ float h[DDIM], t1[DDIM], mu[DDIM], sl[DDIM], nx[XDIM];
  __shared__ float alin[MIXN], lc[MIXN];
  const int t = threadIdx.x;
  const int wave = t >> 5;
  for (int j = t; j < DDIM; j += 256) {
    float s = 0.f;
    for (int b = 0; b < NB; ++b) s += partials[(size_t)b * DDIM + j];
    h[j] = 1.f / (1.f + __expf(-s));
  }
  if (t < XDIM) nx[t] = X[(size_t)(WINN - 1) * XDIM + t];
  __syncthreads();
  for (int tile = wave; tile < DDIM / 16; tile += 8)
    wave_matvec16(Wmu, DDIM, tile * 16, DDIM, h, t1);
  for (int tile = wave; tile < DDIM / 16; tile += 8)
    wave_matvec16(Wsig, DDIM, tile * 16, DDIM, h, sl);
  __syncthreads();
  for (int j = t; j < DDIM; j += 256) {
    t1[j] += bmu[j];
    sl[j] += bsig[j];
  }
  __syncthreads();
  for (int tile = wave; tile < DDIM / 16; tile += 8)
    wave_matvec16(Wmu2, DDIM, tile * 16, DDIM, t1, mu);
  __syncthreads();
  for (int j = t; j < DDIM; j += 256) mu[j] += bmu2[j];
  __syncthreads();
  if (t < MIXN) {
    const int m = t;
    float a = bal[m];
    const float* wr = Wal + (size_t)m * DDIM;
    for (int d = 0; d < DDIM; ++d) a = fmaf(wr[d], h[d], a);
    alin[m] = a;
    float sz = 0.f, ss = 0.f;
    for (int k = 0; k < XDIM; ++k) {
      const float s = sl[m * XDIM + k];           // log(sig)
      const float z = (nx[k] - mu[m * XDIM + k]) * __expf(-s);
      sz = fmaf(z, z, sz);
      ss += s;
    }
    lc[m] = -0.5f * sz - ss - 32.f * 1.8378770664093453f;  // 0.5*XD*log(2pi)
  }
  __syncthreads();
  if (t == 0) {
    float mx = -1e30f;
    for (int m = 0; m < MIXN; ++m) mx = fmaxf(mx, alin[m]);
    float se = 0.f;
    for (int m = 0; m < MIXN; ++m) se += __expf(alin[m] - mx);
    const float lse = mx + __logf(se);
    float v[MIXN];
    float mx2 = -1e30f;
    for (int m = 0; m < MIXN; ++m) {
      v[m] = alin[m] - lse + lc[m];
      mx2 = fmaxf(mx2, v[m]);
    }
    float s2 = 0.f;
    for (int m = 0; m < MIXN; ++m) s2 += __expf(v[m] - mx2);
    out[0] = mx2 + __logf(s2);
  }
  for (int j = t; j < DDIM; j += 256) out[1 + j] = h[j];
}

extern "C" void kernel_launch(void* const* d_in, const int* in_sizes, int n_in,
                              void* d_out, int out_size, void* d_ws, size_t ws_size,
                              hipStream_t stream) {
  const float* X    = (const float*)d_in[0];
  const float* prev = (const float*)d_in[1];
  const float* A    = (const float*)d_in[2];
  const float* W1   = (const float*)d_in[3];
  const float* b1   = (const float*)d_in[4];
  const float* W2   = (const float*)d_in[5];
  const float* b2   = (const float*)d_in[6];
  const float* Wmu  = (const float*)d_in[7];
  const float* bmu  = (const float*)d_in[8];
  const float* Wmu2 = (const float*)d_in[9];
  const float* bmu2 = (const float*)d_in[10];
  const float* Wsig = (const float*)d_in[11];
  const float* bsig = (const float*)d_in[12];
  const float* Wal  = (const float*)d_in[13];
  const float* bal  = (const float*)d_in[14];
  float* out = (float*)d_out;

  // Workspace layout: [0,512) = e vector; [512, 512 + NB*512) = block partials.
  float* evec = (float*)d_ws;
  const size_t availF = ws_size / sizeof(float);
  int NB = 1024;
  while (NB > 64 && (size_t)DDIM + (size_t)NB * DDIM > availF) NB >>= 1;
  const int rowsPerBlock = (RDIM * DDIM) / NB;   // <= 4096
  float* partials = evec + DDIM;

  k_prep<<<1, 256, 0, stream>>>(X, W1, b1, W2, b2, evec);
  k_reduce<<<NB, 256, 0, stream>>>(A, prev, evec, partials, rowsPerBlock);
  k_final<<<1, 256, 0, stream>>>(X, partials, NB, Wmu, bmu, Wmu2, bmu2,
                                 Wsig, bsig, Wal, bal, out);
}